// SequenceEncoder_43241730737030
// MI455X (gfx1250) — compile-verified
//
#include <hip/hip_runtime.h>
#include <hip/hip_bf16.h>
#include <stdint.h>

#define BS 64
#define G  64
#define LSEQ 100
#define D  64
#define GPB 16            // groups per block (one WMMA M-tile)
#define THREADS 256       // 8 wave32s
#define TSTR 68           // padded LDS row stride in floats (272 B, 16B aligned)

typedef __attribute__((ext_vector_type(2))) float v2f;
typedef __attribute__((ext_vector_type(8))) float v8f;

// ---------------------------------------------------------------------------
// Kernel 1: q2 = W2 @ p   (q2[d] = sum_e W2[d][e] * p[e])
// softmax is shift-invariant along l, so attn = softmax(seqs[l] . (W2@p));
// V_last/V_avg/b terms cancel.
// ---------------------------------------------------------------------------
__global__ void precompute_q2_kernel(const float* __restrict__ W2,
                                     const float* __restrict__ p,
                                     float* __restrict__ q2) {
    int d = threadIdx.x;
    if (d < D) {
        float acc = 0.f;
        #pragma unroll 8
        for (int e = 0; e < D; ++e) acc += W2[d * D + e] * p[e];
        q2[d] = acc;
    }
}

// ---------------------------------------------------------------------------
// Kernel 2: attention pooling + MLP head, 16 groups per block.
// Phase 1 (per wave, 2 groups each):
//   - async-stage group tile (100x64 f32) HBM -> LDS (coalesced, single read)
//   - e[l] = seqs[l].q2 -> softmax -> pooled vector into sX
// Phase 2 (8 waves = 8 N-tiles of 16):
//   H0 = X@Wq ; H1 = relu(H0@Wl0+b0) ; H2 = relu(H1@Wl1+b1) ; out = H0+H2
//   via V_WMMA_F32_16X16X4_F32 chains.
// ---------------------------------------------------------------------------
__global__ __launch_bounds__(THREADS) void seq_encoder_main(
    const float* __restrict__ seqs,   // [4096][100][64]
    const float* __restrict__ q2,     // [64]
    const float* __restrict__ Wq,     // [64][128]
    const float* __restrict__ Wl0,    // [128][128]
    const float* __restrict__ bl0,    // [128]
    const float* __restrict__ Wl1,    // [128][128]
    const float* __restrict__ bl1,    // [128]
    float* __restrict__ out)          // [4096][128]
{
    __shared__ float sq2[D];
    __shared__ float sX [GPB][D + 4];     // pooled vectors (padded)
    __shared__ float sH0[GPB][128 + 4];
    __shared__ float sH1[GPB][128 + 4];
    extern __shared__ float smem[];       // 8 waves * LSEQ * TSTR floats

    const int tid  = threadIdx.x;
    const int wave = tid >> 5;            // 0..7
    const int lane = tid & 31;
    const int g0   = blockIdx.x * GPB;    // first group of this block

    float* __restrict__ wtile = smem + wave * (LSEQ * TSTR);  // per-wave tile

    if (tid < D) sq2[tid] = q2[tid];
    __syncthreads();

    // ---------------- Phase 1: attention pooling ----------------
    for (int gsub = 0; gsub < 2; ++gsub) {
        const int gl = wave * 2 + gsub;                       // 0..15
        const float* __restrict__ sp = seqs + (long)(g0 + gl) * (LSEQ * D);

        // --- async stage: HBM -> LDS, 2 rows (512B) per wave instruction ---
        {
            const int sub = lane >> 4;          // row parity within pair
            const int c4  = (lane & 15) * 4;    // float offset (16B chunks)
            // make sure prior-iteration LDS reads have drained before overwrite
            asm volatile("s_wait_dscnt 0x0" ::: "memory");
            #pragma unroll 5
            for (int j = 0; j < LSEQ / 2; ++j) {
                const int row = 2 * j + sub;
                const float* gp = sp + row * D + c4;
                unsigned laddr = (unsigned)(uintptr_t)(wtile + row * TSTR + c4);
                asm volatile("global_load_async_to_lds_b128 %0, %1, off"
                             :: "v"(laddr), "v"(gp) : "memory");
            }
            asm volatile("s_wait_asynccnt 0x0" ::: "memory");
        }

        // --- e[l] = row . q2 ; lane owns rows lane, lane+32, lane+64, lane+96
        float e[4];
        #pragma unroll
        for (int r = 0; r < 4; ++r) {
            const int l = lane + 32 * r;
            if (l < LSEQ) {
                const float4* rp = (const float4*)(wtile + l * TSTR);
                float acc = 0.f;
                #pragma unroll
                for (int i = 0; i < D / 4; ++i) {
                    float4 v = rp[i];
                    acc += v.x * sq2[4*i+0] + v.y * sq2[4*i+1]
                         + v.z * sq2[4*i+2] + v.w * sq2[4*i+3];
                }
                e[r] = acc;
            } else {
                e[r] = -__builtin_huge_valf();
            }
        }
        // wave-wide max
        float m = fmaxf(fmaxf(e[0], e[1]), fmaxf(e[2], e[3]));
        #pragma unroll
        for (int off = 16; off > 0; off >>= 1)
            m = fmaxf(m, __shfl_xor(m, off, 32));
        // exp + wave-wide sum
        float w[4];
        float ssum = 0.f;
        #pragma unroll
        for (int r = 0; r < 4; ++r) {
            w[r] = (lane + 32 * r < LSEQ) ? __expf(e[r] - m) : 0.f;
            ssum += w[r];
        }
        #pragma unroll
        for (int off = 16; off > 0; off >>= 1)
            ssum += __shfl_xor(ssum, off, 32);
        const float inv = 1.0f / ssum;
        #pragma unroll
        for (int r = 0; r < 4; ++r) w[r] *= inv;

        // --- pooled[d] = sum_l attn[l]*row[d]; lane owns d = 2*lane, 2*lane+1
        float ax = 0.f, ay = 0.f;
        #pragma unroll
        for (int r = 0; r < 4; ++r) {
            const int lcount = (r < 3) ? 32 : (LSEQ - 96);
            for (int j = 0; j < lcount; ++j) {
                const float a = __shfl(w[r], j, 32);          // broadcast attn[l]
                const float2 v = *(const float2*)(wtile + (32 * r + j) * TSTR + 2 * lane);
                ax += a * v.x;
                ay += a * v.y;
            }
        }
        sX[gl][2 * lane]     = ax;
        sX[gl][2 * lane + 1] = ay;
    }
    __syncthreads();

    // ---------------- Phase 2: WMMA MLP head ----------------
    // A-frag (16x4 f32): lanes 0-15 -> M=lane, K={k0,k0+1}; lanes 16-31 -> K={k0+2,k0+3}
    // C/D (16x16 f32):   vgpr r, lanes 0-15 -> M=r, N=lane; lanes 16-31 -> M=r+8
    const int row   = lane & 15;
    const int khalf = lane >> 4;          // 0 or 1
    const int col   = 16 * wave + row;    // this wave's N-tile column

    // H0 = X @ Wq  (K = 64)
    v8f c0 = {0.f,0.f,0.f,0.f,0.f,0.f,0.f,0.f};
    #pragma unroll
    for (int k0 = 0; k0 < D; k0 += 4) {
        const int k = k0 + 2 * khalf;
        v2f a, b;
        a.x = sX[row][k];
        a.y = sX[row][k + 1];
        b.x = Wq[k * 128 + col];
        b.y = Wq[(k + 1) * 128 + col];
        c0 = __builtin_amdgcn_wmma_f32_16x16x4_f32(false, a, false, b,
                                                   (short)0, c0, false, false);
    }
    #pragma unroll
    for (int r = 0; r < 8; ++r)
        sH0[r + 8 * khalf][col] = c0[r];
    __syncthreads();

    // H1 = relu(H0 @ Wl0 + bl0)  (K = 128)
    v8f c1 = {0.f,0.f,0.f,0.f,0.f,0.f,0.f,0.f};
    #pragma unroll
    for (int k0 = 0; k0 < 128; k0 += 4) {
        const int k = k0 + 2 * khalf;
        v2f a, b;
        a.x = sH0[row][k];
        a.y = sH0[row][k + 1];
        b.x = Wl0[k * 128 + col];
        b.y = Wl0[(k + 1) * 128 + col];
        c1 = __builtin_amdgcn_wmma_f32_16x16x4_f32(false, a, false, b,
                                                   (short)0, c1, false, false);
    }
    {
        const float bias = bl0[col];
        #pragma unroll
        for (int r = 0; r < 8; ++r)
            sH1[r + 8 * khalf][col] = fmaxf(c1[r] + bias, 0.f);
    }
    __syncthreads();

    // H2 = relu(H1 @ Wl1 + bl1); out = H0 + H2
    v8f c2 = {0.f,0.f,0.f,0.f,0.f,0.f,0.f,0.f};
    #pragma unroll
    for (int k0 = 0; k0 < 128; k0 += 4) {
        const int k = k0 + 2 * khalf;
        v2f a, b;
        a.x = sH1[row][k];
        a.y = sH1[row][k + 1];
        b.x = Wl1[k * 128 + col];
        b.y = Wl1[(k + 1) * 128 + col];
        c2 = __builtin_amdgcn_wmma_f32_16x16x4_f32(false, a, false, b,
                                                   (short)0, c2, false, false);
    }
    {
        const float bias = bl1[col];
        #pragma unroll
        for (int r = 0; r < 8; ++r) {
            const int m = r + 8 * khalf;
            const float h = fmaxf(c2[r] + bias, 0.f);
            out[(long)(g0 + m) * 128 + col] = sH0[m][col] + h;
        }
    }
}

// ---------------------------------------------------------------------------
// Host-side launcher
// ---------------------------------------------------------------------------
extern "C" void kernel_launch(void* const* d_in, const int* in_sizes, int n_in,
                              void* d_out, int out_size, void* d_ws, size_t ws_size,
                              hipStream_t stream) {
    // setup_inputs order:
    // 0 seqs, 1 lens, 2 W1, 3 W2, 4 W3, 5 b, 6 p, 7 Wq, 8 Wl_0, 9 bl_0, 10 Wl_1, 11 bl_1
    const float* seqs = (const float*)d_in[0];
    const float* W2   = (const float*)d_in[3];
    const float* p    = (const float*)d_in[6];
    const float* Wq   = (const float*)d_in[7];
    const float* Wl0  = (const float*)d_in[8];
    const float* bl0  = (const float*)d_in[9];
    const float* Wl1  = (const float*)d_in[10];
    const float* bl1  = (const float*)d_in[11];
    float* out = (float*)d_out;
    float* q2  = (float*)d_ws;   // 64 floats of scratch

    hipLaunchKernelGGL(precompute_q2_kernel, dim3(1), dim3(64), 0, stream, W2, p, q2);

    const int ngroups = BS * G;                                   // 4096
    const size_t dyn_lds = (size_t)(THREADS / 32) * LSEQ * TSTR * sizeof(float); // 217.6 KB
    hipLaunchKernelGGL(seq_encoder_main, dim3(ngroups / GPB), dim3(THREADS), dyn_lds, stream,
                       seqs, q2, Wq, Wl0, bl0, Wl1, bl1, out);
}